// RGCN_85341000172301
// MI455X (gfx1250) — compile-verified
//
#include <hip/hip_runtime.h>

#define D 64
#define TM 32   // edges (or nodes) per wave: two 16-row WMMA tiles sharing B fragments

typedef __attribute__((ext_vector_type(2))) float v2f;
typedef __attribute__((ext_vector_type(8))) float v8f;

// ---------------------------------------------------------------------------
// Small utility kernels
// ---------------------------------------------------------------------------
__global__ void fill_u32_kernel(unsigned* __restrict__ p, unsigned v, int n) {
    int i = blockIdx.x * blockDim.x + threadIdx.x;
    if (i < n) p[i] = v;
}

// degree count per (dst, relation) pair
__global__ void edge_count_kernel(const int* __restrict__ dst, const int* __restrict__ et,
                                  unsigned* __restrict__ cnt, int E, int R) {
    int e = blockIdx.x * blockDim.x + threadIdx.x;
    if (e < E) atomicAdd(&cnt[(size_t)dst[e] * R + et[e]], 1u);
}

// per-edge mean-normalization factor
__global__ void edge_norm_kernel(const int* __restrict__ dst, const int* __restrict__ et,
                                 const unsigned* __restrict__ cnt, float* __restrict__ norm,
                                 int E, int R) {
    int e = blockIdx.x * blockDim.x + threadIdx.x;
    if (e < E) {
        unsigned c = cnt[(size_t)dst[e] * R + et[e]];
        norm[e] = 1.0f / (float)(c > 0u ? c : 1u);
    }
}

__global__ void hist_kernel(const int* __restrict__ et, unsigned* __restrict__ hist, int E) {
    int e = blockIdx.x * blockDim.x + threadIdx.x;
    if (e < E) atomicAdd(&hist[et[e]], 1u);
}

// padded prefix sum over R=65 relations (serial, trivial)
__global__ void offsets_kernel(const unsigned* __restrict__ hist, int* __restrict__ off, int R) {
    if (threadIdx.x == 0 && blockIdx.x == 0) {
        int acc = 0;
        for (int r = 0; r < R; ++r) {
            off[r] = acc;
            acc += (int)((hist[r] + (TM - 1)) / TM) * TM;   // pad each relation to TM
        }
        off[R] = acc;
    }
}

__global__ void scatter_kernel(const int* __restrict__ et, const int* __restrict__ off,
                               unsigned* __restrict__ cursor, int* __restrict__ sorted, int E) {
    int e = blockIdx.x * blockDim.x + threadIdx.x;
    if (e < E) {
        int r = et[e];
        int pos = off[r] + (int)atomicAdd(&cursor[r], 1u);
        sorted[pos] = e;
    }
}

// map each TM-edge tile -> its (single) relation; -1 past the end
__global__ void tile_rel_kernel(const int* __restrict__ off, int* __restrict__ tile_rel,
                                int maxTiles, int R) {
    int t = blockIdx.x * blockDim.x + threadIdx.x;
    if (t >= maxTiles) return;
    int pos = t * TM;
    if (pos >= off[R]) { tile_rel[t] = -1; return; }
    int r = 0;
    while (pos >= off[r + 1]) ++r;   // R=65, L2-hot, trivial
    tile_rel[t] = r;
}

// ---------------------------------------------------------------------------
// Edge-message kernel: one wave32 per 32-edge tile (single relation).
// C(32x64) = A(32x64, gathered src rows) x B(64x64, W[rel]) as two 16-row
// WMMA tiles sharing every B fragment: 16 K-steps x 4 N-tiles x 2 M-tiles
// of v_wmma_f32_16x16x4_f32, then scale by norm and atomic-scatter into agg.
// ---------------------------------------------------------------------------
__global__ __launch_bounds__(32)
void rgcn_msg_wmma_kernel(const float* __restrict__ x, const float* __restrict__ W,
                          const int* __restrict__ src, const int* __restrict__ dst,
                          const float* __restrict__ norm,
                          const int* __restrict__ sorted, const int* __restrict__ tile_rel,
                          float* __restrict__ agg) {
    int t = blockIdx.x;
    int r = tile_rel[t];
    if (r < 0) return;

    __shared__ int   s_src[TM];
    __shared__ int   s_dst[TM];
    __shared__ float s_nrm[TM];

    int lane = threadIdx.x;
    {
        int e = sorted[t * TM + lane];
        if (e >= 0) { s_src[lane] = src[e]; s_dst[lane] = dst[e]; s_nrm[lane] = norm[e]; }
        else        { s_src[lane] = 0;      s_dst[lane] = 0;      s_nrm[lane] = 0.0f;   }
    }
    __syncthreads();   // single-wave workgroup: lowers to S_NOP

    int row  = lane & 15;    // M row within tile (A) / N col (B,C)
    int half = lane >> 4;

    const float* xrow0 = x + (size_t)s_src[row] * D;        // M-tile 0: edges 0-15
    const float* xrow1 = x + (size_t)s_src[16 + row] * D;   // M-tile 1: edges 16-31
    const float* Wr    = W + (size_t)r * D * D;

    v8f acc[2][4];
#pragma unroll
    for (int mt = 0; mt < 2; ++mt)
#pragma unroll
        for (int nt = 0; nt < 4; ++nt) acc[mt][nt] = (v8f){};

#pragma unroll
    for (int k = 0; k < 16; ++k) {
        int kb = 4 * k + 2 * half;       // A: lanes 0-15 K={kb,kb+1}; lanes 16-31 K shifted by +2
        v2f a0; a0[0] = xrow0[kb]; a0[1] = xrow0[kb + 1];
        v2f a1; a1[0] = xrow1[kb]; a1[1] = xrow1[kb + 1];

        const float* w0 = Wr + (size_t)kb * D + row;   // B[kb  ][col] (coalesced per half-wave)
        const float* w1 = w0 + D;                      // B[kb+1][col]
#pragma unroll
        for (int nt = 0; nt < 4; ++nt) {
            v2f b; b[0] = w0[nt * 16]; b[1] = w1[nt * 16];
            acc[0][nt] = __builtin_amdgcn_wmma_f32_16x16x4_f32(false, a0, false, b, (short)0,
                                                               acc[0][nt], false, false);
            acc[1][nt] = __builtin_amdgcn_wmma_f32_16x16x4_f32(false, a1, false, b, (short)0,
                                                               acc[1][nt], false, false);
        }
    }

    // D layout: VGPR i holds M = i + 8*half, N = row (+ 16*ntile)
#pragma unroll
    for (int mt = 0; mt < 2; ++mt) {
#pragma unroll
        for (int i = 0; i < 8; ++i) {
            int m = 16 * mt + i + 8 * half;
            float nm = s_nrm[m];                        // 0.0f for padding rows -> adds exactly +0
            float* drow = agg + (size_t)s_dst[m] * D + row;
            atomicAdd(drow + 0,  acc[mt][0][i] * nm);
            atomicAdd(drow + 16, acc[mt][1][i] * nm);
            atomicAdd(drow + 32, acc[mt][2][i] * nm);
            atomicAdd(drow + 48, acc[mt][3][i] * nm);
        }
    }
}

// ---------------------------------------------------------------------------
// Self-loop kernel: out = relu(agg + x @ Wl). One wave32 per 32 nodes.
// ---------------------------------------------------------------------------
__global__ __launch_bounds__(32)
void rgcn_selfloop_wmma_kernel(const float* __restrict__ x, const float* __restrict__ Wl,
                               const float* __restrict__ agg, float* __restrict__ out, int Nn) {
    int base = blockIdx.x * TM;
    if (base >= Nn) return;
    int lane = threadIdx.x;
    int row  = lane & 15;
    int half = lane >> 4;

    int node0 = base + row;        if (node0 >= Nn) node0 = Nn - 1;   // clamped; output guarded
    int node1 = base + 16 + row;   if (node1 >= Nn) node1 = Nn - 1;
    const float* xrow0 = x + (size_t)node0 * D;
    const float* xrow1 = x + (size_t)node1 * D;

    v8f acc[2][4];
#pragma unroll
    for (int mt = 0; mt < 2; ++mt)
#pragma unroll
        for (int nt = 0; nt < 4; ++nt) acc[mt][nt] = (v8f){};

#pragma unroll
    for (int k = 0; k < 16; ++k) {
        int kb = 4 * k + 2 * half;
        v2f a0; a0[0] = xrow0[kb]; a0[1] = xrow0[kb + 1];
        v2f a1; a1[0] = xrow1[kb]; a1[1] = xrow1[kb + 1];
        const float* w0 = Wl + (size_t)kb * D + row;
        const float* w1 = w0 + D;
#pragma unroll
        for (int nt = 0; nt < 4; ++nt) {
            v2f b; b[0] = w0[nt * 16]; b[1] = w1[nt * 16];
            acc[0][nt] = __builtin_amdgcn_wmma_f32_16x16x4_f32(false, a0, false, b, (short)0,
                                                               acc[0][nt], false, false);
            acc[1][nt] = __builtin_amdgcn_wmma_f32_16x16x4_f32(false, a1, false, b, (short)0,
                                                               acc[1][nt], false, false);
        }
    }

#pragma unroll
    for (int mt = 0; mt < 2; ++mt) {
#pragma unroll
        for (int i = 0; i < 8; ++i) {
            int nm = base + 16 * mt + i + 8 * half;
            if (nm < Nn) {
                const float* arow = agg + (size_t)nm * D + row;
                float*       orow = out + (size_t)nm * D + row;
                orow[0]  = fmaxf(acc[mt][0][i] + arow[0],  0.0f);
                orow[16] = fmaxf(acc[mt][1][i] + arow[16], 0.0f);
                orow[32] = fmaxf(acc[mt][2][i] + arow[32], 0.0f);
                orow[48] = fmaxf(acc[mt][3][i] + arow[48], 0.0f);
            }
        }
    }
}

// ---------------------------------------------------------------------------
// Host-side orchestration
// ---------------------------------------------------------------------------
extern "C" void kernel_launch(void* const* d_in, const int* in_sizes, int n_in,
                              void* d_out, int out_size, void* d_ws, size_t ws_size,
                              hipStream_t stream) {
    const float* x   = (const float*)d_in[0];
    const int*   src = (const int*)d_in[1];
    const int*   dst = (const int*)d_in[2];
    const int*   et  = (const int*)d_in[3];
    const float* W1  = (const float*)d_in[4];
    const float* Wl1 = (const float*)d_in[5];
    const float* W2  = (const float*)d_in[6];
    const float* Wl2 = (const float*)d_in[7];
    float* out = (float*)d_out;

    const int Nn = in_sizes[0] / D;        // 50000
    const int E  = in_sizes[1];            // 800000
    const int R  = in_sizes[4] / (D * D);  // 65

    const int padLen   = ((E + TM - 1) / TM) * TM + TM * R;  // worst-case padded edge list
    const int maxTiles = padLen / TM;

    // carve scratch (256B aligned) out of d_ws: ~45 MB total
    char* p = (char*)d_ws;
    auto carve = [&](size_t bytes) -> void* {
        void* q = (void*)p;
        p += (bytes + 255) & ~(size_t)255;
        return q;
    };
    unsigned* cnt    = (unsigned*)carve((size_t)Nn * R * 4);
    float*    norm   = (float*)   carve((size_t)E * 4);
    int*      sorted = (int*)     carve((size_t)padLen * 4);
    int*      trel   = (int*)     carve((size_t)maxTiles * 4);
    unsigned* hist   = (unsigned*)carve((size_t)R * 4);
    int*      off    = (int*)     carve((size_t)(R + 1) * 4);
    unsigned* cursor = (unsigned*)carve((size_t)R * 4);
    float*    agg    = (float*)   carve((size_t)Nn * D * 4);
    float*    h      = (float*)   carve((size_t)Nn * D * 4);
    (void)ws_size; (void)n_in; (void)out_size;

    const int TB = 256;
    auto nb = [](long n, int tb) { return (int)((n + tb - 1) / tb); };

    // --- graph preprocessing (shared by both layers) ---
    fill_u32_kernel<<<nb((long)Nn * R, TB), TB, 0, stream>>>(cnt, 0u, Nn * R);
    edge_count_kernel<<<nb(E, TB), TB, 0, stream>>>(dst, et, cnt, E, R);
    edge_norm_kernel <<<nb(E, TB), TB, 0, stream>>>(dst, et, cnt, norm, E, R);
    fill_u32_kernel<<<nb(R, TB), TB, 0, stream>>>(hist, 0u, R);
    hist_kernel<<<nb(E, TB), TB, 0, stream>>>(et, hist, E);
    offsets_kernel<<<1, 32, 0, stream>>>(hist, off, R);
    fill_u32_kernel<<<nb(padLen, TB), TB, 0, stream>>>((unsigned*)sorted, 0xFFFFFFFFu, padLen);
    fill_u32_kernel<<<nb(R, TB), TB, 0, stream>>>(cursor, 0u, R);
    scatter_kernel<<<nb(E, TB), TB, 0, stream>>>(et, off, cursor, sorted, E);
    tile_rel_kernel<<<nb(maxTiles, TB), TB, 0, stream>>>(off, trel, maxTiles, R);

    const int nodeTiles = (Nn + TM - 1) / TM;

    // --- layer 1 ---
    fill_u32_kernel<<<nb((long)Nn * D, TB), TB, 0, stream>>>((unsigned*)agg, 0u, Nn * D);
    rgcn_msg_wmma_kernel<<<maxTiles, 32, 0, stream>>>(x, W1, src, dst, norm, sorted, trel, agg);
    rgcn_selfloop_wmma_kernel<<<nodeTiles, 32, 0, stream>>>(x, Wl1, agg, h, Nn);

    // --- layer 2 ---
    fill_u32_kernel<<<nb((long)Nn * D, TB), TB, 0, stream>>>((unsigned*)agg, 0u, Nn * D);
    rgcn_msg_wmma_kernel<<<maxTiles, 32, 0, stream>>>(h, W2, src, dst, norm, sorted, trel, agg);
    rgcn_selfloop_wmma_kernel<<<nodeTiles, 32, 0, stream>>>(h, Wl2, agg, out, Nn);
}